// NewCombinedLoss_1554778161209
// MI455X (gfx1250) — compile-verified
//
#include <hip/hip_runtime.h>
#include <hip/hip_bf16.h>
#include <math.h>

typedef float v2f __attribute__((ext_vector_type(2)));
typedef float v8f __attribute__((ext_vector_type(8)));

#define NV      (64*64*64)      // voxels per (b) volume
#define BATCH   4
#define NCLS    4
#define BIGF    1e8f
#define SMOOTHF 1e-5f
#define LN2F    0.69314718055994530942f

// accumulator layout (floats) at start of d_ws
#define ACC_INTER  0    // 16: diag of P x O^T  -> sum probs*onehot per (b,c)
#define ACC_PSUM   16   // 16: row sums of P    -> sum probs per (b,c)
#define ACC_CNT    32   // 16: col sums of O    -> sum onehot per (b,c)
#define ACC_BDOT   48   // 16: sum sdf*probs per (b,c)
#define ACC_CE     64   // 1 : sum of -log_softmax[target]
#define ACC_FLOATS 80

#define GRID1   512                      // blocks for fused stats kernel
#define WAVES1  (GRID1*256/32)           // 4096 waves
#define CHUNK1  (NV / WAVES1)            // 64 voxels per wave (multiple of 4)

// ---------------------------------------------------------------------------
// Kernel 1: softmax + CE + dice statistics via V_WMMA_F32_16X16X4_F32.
// Rows/cols m,n = 4*b + c (exactly 16 (b,c) pairs). K dimension = voxels.
//   C_inter += A(probs) x B(onehot)   -> diagonal = dice intersections
//   C_rows  += A(probs) x ones        -> row m    = sum of probs (b,c)
//   C_cols  += ones     x B(onehot)   -> col n    = one-hot counts (b,c)
// A fragment (16x4 f32): lane l holds row M=l%16, K = 2*(l>=16)+v  (v=0,1).
// B fragment mirrors (col N=l%16, same K), so each lane needs probs/onehot
// for ITS (b,c) at voxels vb+k0, vb+k0+1 only.
// Loop body is fully branchless: the target-logit pick is a flat sum of
// independently gated terms (v_cmp + v_cndmask each, no nest to branchify),
// CE is gated by a select, normalization uses v_rcp_f32, and log(s) uses raw
// v_log_f32 * ln2 (s is in [1,4], no fixup needed).
// ---------------------------------------------------------------------------
__global__ void fused_stats_kernel(const float* __restrict__ preds,
                                   const int*   __restrict__ targets,
                                   float* __restrict__ acc) {
  const int lane = threadIdx.x & 31;
  const int wave = (blockIdx.x * blockDim.x + threadIdx.x) >> 5;
  const int m    = lane & 15;
  const int b    = m >> 2;
  const int cc   = m & 3;
  const int k0   = (lane >> 4) << 1;          // 0 or 2
  const int start = wave * CHUNK1;
  const float ce_gate = (cc == 0) ? 1.0f : 0.0f;

  v8f c_inter = {}; v8f c_rows = {}; v8f c_cols = {};
  v2f ones2; ones2.x = 1.0f; ones2.y = 1.0f;
  float ce = 0.0f;

  const float* pb = preds   + (size_t)b * NCLS * NV;
  const int*   tb = targets + (size_t)b * NV;

  for (int vb = start; vb < start + CHUNK1; vb += 4) {
    v2f a, bm;
#pragma unroll
    for (int t = 0; t < 2; ++t) {
      const int vox = vb + k0 + t;
      const float l0 = pb[0 * NV + vox];
      const float l1 = pb[1 * NV + vox];
      const float l2 = pb[2 * NV + vox];
      const float l3 = pb[3 * NV + vox];
      const float mx = fmaxf(fmaxf(l0, l1), fmaxf(l2, l3));
      const float e0 = __expf(l0 - mx), e1 = __expf(l1 - mx);
      const float e2 = __expf(l2 - mx), e3 = __expf(l3 - mx);
      const float s  = e0 + e1 + e2 + e3;
      const float rs = __builtin_amdgcn_rcpf(s);          // v_rcp_f32
      const float ec = (cc == 0) ? e0 : (cc == 1) ? e1 : (cc == 2) ? e2 : e3;
      const int   tg = tb[vox];
      a[t]  = ec * rs;
      bm[t] = (tg == cc) ? 1.0f : 0.0f;
      // branchless target-logit pick: flat gated sum (no nested ternary)
      const float lt = ((tg == 0) ? l0 : 0.0f) + ((tg == 1) ? l1 : 0.0f) +
                       ((tg == 2) ? l2 : 0.0f) + ((tg == 3) ? l3 : 0.0f);
      // lanes {4b, 16+4b} cover every (b,vox) exactly once
      const float logs = __builtin_amdgcn_logf(s) * LN2F;  // s in [1,4]
      ce += ce_gate * (mx + logs - lt);
    }
    c_inter = __builtin_amdgcn_wmma_f32_16x16x4_f32(false, a, false, bm,
                                                    (short)0, c_inter, false, false);
    c_rows  = __builtin_amdgcn_wmma_f32_16x16x4_f32(false, a, false, ones2,
                                                    (short)0, c_rows, false, false);
    c_cols  = __builtin_amdgcn_wmma_f32_16x16x4_f32(false, ones2, false, bm,
                                                    (short)0, c_cols, false, false);
  }

  // D layout: lane l, vgpr v -> M = v + 8*(l>=16), N = l%16.
  // Diagonal: lanes 0..7 at d[l]; lanes 24..31 at d[l-24].
#pragma unroll
  for (int v = 0; v < 8; ++v) {
    if (lane == v)      atomicAdd(&acc[ACC_INTER + v],     c_inter[v]);
    if (lane == 24 + v) atomicAdd(&acc[ACC_INTER + 8 + v], c_inter[v]);
  }
  if (lane == 0) {      // N=0 column: rows 0..7
#pragma unroll
    for (int v = 0; v < 8; ++v) atomicAdd(&acc[ACC_PSUM + v], c_rows[v]);
  }
  if (lane == 16) {     // N=0 column: rows 8..15
#pragma unroll
    for (int v = 0; v < 8; ++v) atomicAdd(&acc[ACC_PSUM + 8 + v], c_rows[v]);
  }
  if (lane < 16) atomicAdd(&acc[ACC_CNT + lane], c_cols[0]);  // row M=0, N=lane
  if (cc == 0)   atomicAdd(&acc[ACC_CE], ce);
}

// ---------------------------------------------------------------------------
// EDT init: field[0][b][v] = mask ? 0 : BIG  (distance-to-mask)
//           field[1][b][v] = mask ? BIG : 0  (distance-to-~mask)
// ---------------------------------------------------------------------------
__global__ void edt_init_kernel(const int* __restrict__ targets,
                                float* __restrict__ out, int cls) {
  const int idx = blockIdx.x * 256 + threadIdx.x;          // < 2*BATCH*NV
  const int FV  = BATCH * NV;
  const int fi  = (idx >= FV) ? 1 : 0;
  const int v   = idx - fi * FV;
  const bool mask = (targets[v] == cls);
  const bool zero = fi ? !mask : mask;
  out[idx] = zero ? 0.0f : BIGF;
}

// ---------------------------------------------------------------------------
// One exact 1D min-plus pass: out[i] = min_j in[j] + (i-j)^2 along `stride`
// axis (n = 64). 4 lines per 256-thread block, line staged in LDS, inner loop
// is v_fma_f32 + v_min_num_f32 per j.
// ---------------------------------------------------------------------------
__global__ void edt_pass_kernel(const float* __restrict__ in,
                                float* __restrict__ out, int stride) {
  __shared__ float lds[4][64];
  const int lline = threadIdx.x >> 6;          // 0..3
  const int i     = threadIdx.x & 63;
  const int L     = blockIdx.x * 4 + lline;    // global line id
  const int hi    = L / stride;
  const int lo    = L - hi * stride;
  const int base  = hi * (64 * stride) + lo;

  lds[lline][i] = in[base + i * stride];
  __syncthreads();

  const float fi = (float)i;
  float best = 3.0e38f;
  float jf = 0.0f;
#pragma unroll 8
  for (int j = 0; j < 64; ++j) {
    const float d = fi - jf;
    best = fminf(best, __builtin_fmaf(d, d, lds[lline][j]));
    jf += 1.0f;
  }
  out[base + i * stride] = best;
}

// ---------------------------------------------------------------------------
// Boundary dot: sum over voxels of sdf * probs[:,cls]; sdf = sqrt(d_out) -
// sqrt(d_in), overridden to +-1 for empty/full masks (counts from WMMA pass).
// ---------------------------------------------------------------------------
__global__ void boundary_dot_kernel(const float* __restrict__ preds,
                                    const float* __restrict__ dfield,
                                    float* __restrict__ acc, int cls) {
  __shared__ float red[256];
  const int idx = blockIdx.x * 256 + threadIdx.x;          // < BATCH*NV
  const int b   = idx / NV;
  const int v   = idx - b * NV;
  const int FV  = BATCH * NV;

  float sdf = sqrtf(dfield[idx]) - sqrtf(dfield[FV + idx]);
  const float cnt = acc[ACC_CNT + 4 * b + cls];
  if (cnt == 0.0f)            sdf =  1.0f;
  else if (cnt == (float)NV)  sdf = -1.0f;

  const float* pb = preds + (size_t)b * NCLS * NV;
  const float l0 = pb[v], l1 = pb[NV + v], l2 = pb[2 * NV + v], l3 = pb[3 * NV + v];
  const float mx = fmaxf(fmaxf(l0, l1), fmaxf(l2, l3));
  const float e0 = __expf(l0 - mx), e1 = __expf(l1 - mx);
  const float e2 = __expf(l2 - mx), e3 = __expf(l3 - mx);
  const float s  = e0 + e1 + e2 + e3;
  const float ec = (cls == 1) ? e1 : (cls == 2) ? e2 : e3;

  red[threadIdx.x] = sdf * ec * __builtin_amdgcn_rcpf(s);
  __syncthreads();
  for (int off = 128; off > 0; off >>= 1) {
    if (threadIdx.x < off) red[threadIdx.x] += red[threadIdx.x + off];
    __syncthreads();
  }
  if (threadIdx.x == 0) atomicAdd(&acc[ACC_BDOT + 4 * b + cls], red[0]);
}

// ---------------------------------------------------------------------------
// Final scalar combine.
// ---------------------------------------------------------------------------
__global__ void combine_kernel(const float* __restrict__ acc,
                               float* __restrict__ out) {
  if (threadIdx.x == 0 && blockIdx.x == 0) {
    float dsum = 0.0f;
    for (int m = 0; m < 16; ++m)
      dsum += (2.0f * acc[ACC_INTER + m] + SMOOTHF) /
              (acc[ACC_PSUM + m] + acc[ACC_CNT + m] + SMOOTHF);
    const float l_dice = 1.0f - dsum / 16.0f;
    const float l_ce   = acc[ACC_CE] / (float)(BATCH * NV);
    float bsum = 0.0f;
    for (int b = 0; b < BATCH; ++b)
      for (int c = 1; c < NCLS; ++c)
        bsum += acc[ACC_BDOT + 4 * b + c] / (float)NV;
    const float l_bound = bsum / (float)(BATCH * (NCLS - 1));
    out[0] = l_dice + l_ce + 0.01f * l_bound;
  }
}

extern "C" void kernel_launch(void* const* d_in, const int* in_sizes, int n_in,
                              void* d_out, int out_size, void* d_ws, size_t ws_size,
                              hipStream_t stream) {
  const float* preds   = (const float*)d_in[0];
  const int*   targets = (const int*)d_in[1];
  float* acc  = (float*)d_ws;
  float* bufA = (float*)((char*)d_ws + 1024);           // 2*BATCH*NV floats (8 MB)
  float* bufB = bufA + (size_t)2 * BATCH * NV;          // 8 MB more

  hipMemsetAsync(acc, 0, ACC_FLOATS * sizeof(float), stream);

  fused_stats_kernel<<<GRID1, 256, 0, stream>>>(preds, targets, acc);

  const int elems2 = 2 * BATCH * NV;                    // both polarities
  for (int cls = 1; cls < NCLS; ++cls) {
    edt_init_kernel<<<elems2 / 256, 256, 0, stream>>>(targets, bufA, cls);
    edt_pass_kernel<<<elems2 / 256, 256, 0, stream>>>(bufA, bufB, 4096); // D
    edt_pass_kernel<<<elems2 / 256, 256, 0, stream>>>(bufB, bufA, 64);   // H
    edt_pass_kernel<<<elems2 / 256, 256, 0, stream>>>(bufA, bufB, 1);    // W
    boundary_dot_kernel<<<(BATCH * NV) / 256, 256, 0, stream>>>(preds, bufB, acc, cls);
  }

  combine_kernel<<<1, 64, 0, stream>>>(acc, (float*)d_out);
}